// GCNRegressor_33706903339183
// MI455X (gfx1250) — compile-verified
//
#include <hip/hip_runtime.h>
#include <hip/hip_bf16.h>

#define N_NODES 50000
#define N_EDGES 800000
#define HIDDEN 128
#define N_GRAPHS 512
#define N_PAD 50048            // 391 * 128, zero-padded rows for divergence-free WMMA GEMM
#define LN_EPS 1e-5f

typedef __attribute__((ext_vector_type(2))) float v2f;
typedef __attribute__((ext_vector_type(8))) float v8f;

// ---------------- embedding gather (+ zero pad rows) ----------------
__global__ void k_embed(const int* __restrict__ x, const float* __restrict__ emb,
                        float* __restrict__ h0) {
  int tid = blockIdx.x * blockDim.x + threadIdx.x;   // one float4 per thread
  int i4 = tid * 4;
  if (i4 >= N_PAD * 128) return;
  int n = i4 >> 7;
  int d = i4 & 127;
  float4 v = make_float4(0.f, 0.f, 0.f, 0.f);
  if (n < N_NODES) {
    int tok = x[n];
    v = *(const float4*)(emb + (size_t)tok * 128 + d);
  }
  *(float4*)(h0 + i4) = v;
}

// ---------------- degree: init to 1 (self loop), accumulate, rsqrt ----------------
__global__ void k_deg_init(float* __restrict__ deg) {
  int i = blockIdx.x * blockDim.x + threadIdx.x;
  if (i < N_PAD) deg[i] = 1.0f;
}
__global__ void k_deg_acc(const int* __restrict__ dst, float* __restrict__ deg) {
  int e = blockIdx.x * blockDim.x + threadIdx.x;
  if (e < N_EDGES) atomicAdd(&deg[dst[e]], 1.0f);
}
__global__ void k_deg_fin(float* __restrict__ deg) {
  int i = blockIdx.x * blockDim.x + threadIdx.x;
  if (i < N_NODES) deg[i] = rsqrtf(deg[i]);
}

// ---------------- fp32 WMMA GEMM: out[N_PAD,128] = A[N_PAD,128] @ W[128,128] ----------------
// One workgroup = 8 waves = 128 rows. W^T staged in LDS (stride 130 floats keeps
// float2 B-fragments 8B-aligned and spreads banks).
__global__ void __launch_bounds__(256) k_gemm(const float* __restrict__ A,
                                              const float* __restrict__ W,
                                              float* __restrict__ out) {
  __shared__ float wt[128 * 130];
  int tid = threadIdx.x;
#pragma unroll
  for (int it = 0; it < 64; ++it) {           // 16384 floats / 256 threads
    int idx = it * 256 + tid;
    int k = idx >> 7, n = idx & 127;
    wt[n * 130 + k] = W[idx];                 // transpose: wt[n][k] = W[k][n]
  }
  __syncthreads();

  int lane = tid & 31;
  int wave = tid >> 5;                        // 0..7
  int half = lane >> 4;                       // K-half select (K+0/1 vs K+2/3)
  int l16  = lane & 15;                       // M (for A) / N (for B) within tile
  int rowBase = blockIdx.x * 128 + wave * 16;
  const float* arow = A + (size_t)(rowBase + l16) * 128;

  v8f acc[8];
#pragma unroll
  for (int t = 0; t < 8; ++t) acc[t] = (v8f)0.0f;

  for (int c = 0; c < 32; ++c) {              // K in chunks of 4
    int k0 = 4 * c + 2 * half;
    v2f a = *(const v2f*)(arow + k0);         // A frag: rows rowBase..+15, K = k0,k0+1
#pragma unroll
    for (int t = 0; t < 8; ++t) {             // 8 N-tiles of 16
      v2f b = *(const v2f*)(&wt[(16 * t + l16) * 130 + k0]);
      acc[t] = __builtin_amdgcn_wmma_f32_16x16x4_f32(
          false, a, false, b, (short)0, acc[t], false, false);
    }
  }

#pragma unroll
  for (int t = 0; t < 8; ++t) {
#pragma unroll
    for (int r = 0; r < 8; ++r) {             // C/D layout: VGPR r -> M = r + 8*half
      out[(size_t)(rowBase + r + 8 * half) * 128 + 16 * t + l16] = acc[t][r];
    }
  }
}

// ---------------- self-loop init: agg = hw * dinv^2 ----------------
__global__ void k_selfloop(const float* __restrict__ hw, const float* __restrict__ dinv,
                           float* __restrict__ agg) {
  int tid = blockIdx.x * blockDim.x + threadIdx.x;
  int i4 = tid * 4;
  if (i4 >= N_NODES * 128) return;
  int n = i4 >> 7;
  float s = dinv[n];
  s = s * s;
  float4 v = *(const float4*)(hw + i4);
  v.x *= s; v.y *= s; v.z *= s; v.w *= s;
  *(float4*)(agg + i4) = v;
}

// ---------------- edge scatter: 32 lanes per edge, 4 dims per lane ----------------
__global__ void k_scatter(const int* __restrict__ src, const int* __restrict__ dst,
                          const float* __restrict__ dinv, const float* __restrict__ hw,
                          float* __restrict__ agg) {
  int tid = blockIdx.x * blockDim.x + threadIdx.x;
  int e = tid >> 5;
  int lane = tid & 31;
  if (e >= N_EDGES) return;
  int s = src[e], d = dst[e];
  float norm = dinv[s] * dinv[d];
  int off = lane * 4;
  float4 v = *(const float4*)(hw + (size_t)s * 128 + off);
  float* ap = agg + (size_t)d * 128 + off;
  atomicAdd(ap + 0, v.x * norm);
  atomicAdd(ap + 1, v.y * norm);
  atomicAdd(ap + 2, v.z * norm);
  atomicAdd(ap + 3, v.w * norm);
}

// ---------------- fused bias + LayerNorm + ReLU (one wave per node) ----------------
__global__ void k_postln(const float* __restrict__ agg, const float* __restrict__ b,
                         const float* __restrict__ g, const float* __restrict__ be,
                         float* __restrict__ out) {
  int tid = blockIdx.x * blockDim.x + threadIdx.x;
  int node = tid >> 5;
  int lane = tid & 31;
  if (node >= N_NODES) return;
  int d = lane * 4;
  float4 a4 = *(const float4*)(agg + (size_t)node * 128 + d);
  float4 b4 = *(const float4*)(b + d);
  float vx = a4.x + b4.x, vy = a4.y + b4.y, vz = a4.z + b4.z, vw = a4.w + b4.w;
  float s = vx + vy + vz + vw;
#pragma unroll
  for (int m = 16; m >= 1; m >>= 1) s += __shfl_xor(s, m, 32);
  float mu = s * (1.f / 128.f);
  float cx = vx - mu, cy = vy - mu, cz = vz - mu, cw = vw - mu;
  float vs = cx * cx + cy * cy + cz * cz + cw * cw;
#pragma unroll
  for (int m = 16; m >= 1; m >>= 1) vs += __shfl_xor(vs, m, 32);
  float rs = rsqrtf(vs * (1.f / 128.f) + LN_EPS);
  float4 g4 = *(const float4*)(g + d);
  float4 e4 = *(const float4*)(be + d);
  float4 o;
  o.x = fmaxf(0.f, cx * rs * g4.x + e4.x);
  o.y = fmaxf(0.f, cy * rs * g4.y + e4.y);
  o.z = fmaxf(0.f, cz * rs * g4.z + e4.z);
  o.w = fmaxf(0.f, cw * rs * g4.w + e4.w);
  *(float4*)(out + (size_t)node * 128 + d) = o;
}

// ---------------- pooling ----------------
__global__ void k_pool_zero(float* __restrict__ pooled) {
  int i = blockIdx.x * blockDim.x + threadIdx.x;
  if (i < N_GRAPHS * 128 + N_GRAPHS) pooled[i] = 0.0f;   // sums then counts
}
__global__ void k_pool_acc(const float* __restrict__ h, const int* __restrict__ batch,
                           float* __restrict__ pooled, float* __restrict__ counts) {
  int tid = blockIdx.x * blockDim.x + threadIdx.x;
  int n = tid >> 5;
  int lane = tid & 31;
  if (n >= N_NODES) return;
  int gidx = batch[n];
  int d = lane * 4;
  float4 v = *(const float4*)(h + (size_t)n * 128 + d);
  float* pp = pooled + (size_t)gidx * 128 + d;
  atomicAdd(pp + 0, v.x);
  atomicAdd(pp + 1, v.y);
  atomicAdd(pp + 2, v.z);
  atomicAdd(pp + 3, v.w);
  if (lane == 0) atomicAdd(&counts[gidx], 1.0f);
}
__global__ void k_fc(const float* __restrict__ pooled, const float* __restrict__ counts,
                     const float* __restrict__ fcW, const float* __restrict__ fcb,
                     float* __restrict__ out) {
  int tid = blockIdx.x * blockDim.x + threadIdx.x;
  int gidx = tid >> 5;
  int lane = tid & 31;
  if (gidx >= N_GRAPHS) return;
  float inv = 1.0f / fmaxf(counts[gidx], 1.0f);
  int d = lane * 4;
  float4 p = *(const float4*)(pooled + (size_t)gidx * 128 + d);
  float4 w = *(const float4*)(fcW + d);
  float s = (p.x * w.x + p.y * w.y + p.z * w.z + p.w * w.w) * inv;
#pragma unroll
  for (int m = 16; m >= 1; m >>= 1) s += __shfl_xor(s, m, 32);
  if (lane == 0) out[gidx] = s + fcb[0];
}

extern "C" void kernel_launch(void* const* d_in, const int* in_sizes, int n_in,
                              void* d_out, int out_size, void* d_ws, size_t ws_size,
                              hipStream_t stream) {
  (void)in_sizes; (void)n_in; (void)out_size; (void)ws_size;
  const int*   x     = (const int*)d_in[0];
  const int*   ei    = (const int*)d_in[1];        // [2, N_EDGES] row-major
  const int*   batch = (const int*)d_in[2];
  const float* emb   = (const float*)d_in[3];
  const float* W1    = (const float*)d_in[4];
  const float* b1    = (const float*)d_in[5];
  const float* g1    = (const float*)d_in[6];
  const float* be1   = (const float*)d_in[7];
  const float* W2    = (const float*)d_in[8];
  const float* b2    = (const float*)d_in[9];
  const float* g2    = (const float*)d_in[10];
  const float* be2   = (const float*)d_in[11];
  const float* fcW   = (const float*)d_in[12];
  const float* fcb   = (const float*)d_in[13];
  const int* src = ei;
  const int* dst = ei + N_EDGES;
  float* out = (float*)d_out;

  // workspace layout (floats)
  float* ws = (float*)d_ws;
  const size_t FEAT = (size_t)N_PAD * 128;
  float* buf0   = ws;                 // ping
  float* buf1   = buf0 + FEAT;        // pong
  float* dinv   = buf1 + FEAT;        // N_PAD
  float* pooled = dinv + N_PAD;       // N_GRAPHS*128 sums
  float* counts = pooled + (size_t)N_GRAPHS * 128;  // N_GRAPHS

  const int B = 256;
  const int gFeat4   = (N_PAD * 128 / 4 + B - 1) / B;
  const int gSelf4   = (N_NODES * 128 / 4 + B - 1) / B;
  const int gNodes   = (N_PAD + B - 1) / B;
  const int gEdges   = (N_EDGES + B - 1) / B;
  const int gEdge32  = (int)(((size_t)N_EDGES * 32 + B - 1) / B);
  const int gNode32  = (int)(((size_t)N_NODES * 32 + B - 1) / B);
  const int gGemm    = N_PAD / 128;   // 391
  const int gPoolZ   = (N_GRAPHS * 128 + N_GRAPHS + B - 1) / B;
  const int gFc      = (N_GRAPHS * 32 + B - 1) / B;

  // h0 = emb[x] (zero padded rows)
  k_embed<<<gFeat4, B, 0, stream>>>(x, emb, buf0);
  // dinv = rsqrt(deg + 1)
  k_deg_init<<<gNodes, B, 0, stream>>>(dinv);
  k_deg_acc<<<gEdges, B, 0, stream>>>(dst, dinv);
  k_deg_fin<<<gNodes, B, 0, stream>>>(dinv);

  // ---- layer 1 ----
  k_gemm<<<gGemm, B, 0, stream>>>(buf0, W1, buf1);             // buf1 = h0 @ W1
  k_selfloop<<<gSelf4, B, 0, stream>>>(buf1, dinv, buf0);      // buf0 = agg init
  k_scatter<<<gEdge32, B, 0, stream>>>(src, dst, dinv, buf1, buf0);
  k_postln<<<gNode32, B, 0, stream>>>(buf0, b1, g1, be1, buf1); // buf1 = h1 (pad rows stay 0)

  // ---- layer 2 ----
  k_gemm<<<gGemm, B, 0, stream>>>(buf1, W2, buf0);             // buf0 = h1 @ W2
  k_selfloop<<<gSelf4, B, 0, stream>>>(buf0, dinv, buf1);      // buf1 = agg init
  k_scatter<<<gEdge32, B, 0, stream>>>(src, dst, dinv, buf0, buf1);
  k_postln<<<gNode32, B, 0, stream>>>(buf1, b2, g2, be2, buf0); // buf0 = h2

  // ---- mean pool + FC ----
  k_pool_zero<<<gPoolZ, B, 0, stream>>>(pooled);
  k_pool_acc<<<gNode32, B, 0, stream>>>(buf0, batch, pooled, counts);
  k_fc<<<gFc, B, 0, stream>>>(pooled, counts, fcW, fcb, out);
}